// IntegralTransformBatch_4329327034823
// MI455X (gfx1250) — compile-verified
//
#include <hip/hip_runtime.h>
#include <math.h>

// Problem constants (match reference setup_inputs)
#define B_    8
#define N_    9225
#define M_    4096
#define K_    32
#define HID_  64
#define DOUT_ 16

typedef float v2f __attribute__((ext_vector_type(2)));
typedef float v8f __attribute__((ext_vector_type(8)));

// D(16x16,f32) = A(16x4,f32) x B(4x16,f32) + C  — fp32 matrix pipe, wave32.
__device__ __forceinline__ v8f wmma_f32_4(v2f a, v2f b, v8f c) {
    return __builtin_amdgcn_wmma_f32_16x16x4_f32(
        /*neg_a=*/false, a, /*neg_b=*/false, b,
        /*c_mod=*/(short)0, c, /*reuse_a=*/false, /*reuse_b=*/false);
}

// ds_swizzle group-of-32 patterns: new_lane = ((lane & and) | or) ^ xor
// offset[14:10]=xor_mask, offset[9:5]=or_mask, offset[4:0]=and_mask.
// Immediate-encoded lane permute: no VALU address setup (vs __shfl_xor's
// mbcnt/xor/cndmask/lshl sequence feeding ds_bpermute).
#define SWZ_XOR16 0x401f
#define SWZ_XOR8  0x201f
#define SWZ_XOR4  0x101f
#define SWZ_XOR2  0x081f
#define SWZ_XOR1  0x041f

template <int PAT>
__device__ __forceinline__ float swz(float x) {
    return __builtin_bit_cast(
        float, __builtin_amdgcn_ds_swizzle(__builtin_bit_cast(int, x), PAT));
}

// Branch-free exact-erf GELU. Abramowitz–Stegun 7.1.26 rational approximation
// of erf (max abs err 1.5e-7), implemented with raw v_rcp_f32 / v_exp_f32 so
// no exec-mask divergence is generated (keeps EXEC all-ones around WMMA).
__device__ __forceinline__ float gelu_exact(float z) {
    const float s  = z * 0.70710678118654752440f;   // z / sqrt(2)
    const float ax = __builtin_fabsf(s);
    const float t  = __builtin_amdgcn_rcpf(__builtin_fmaf(0.3275911f, ax, 1.0f));
    float p = 1.061405429f;
    p = __builtin_fmaf(p, t, -1.453152027f);
    p = __builtin_fmaf(p, t,  1.421413741f);
    p = __builtin_fmaf(p, t, -0.284496736f);
    p = __builtin_fmaf(p, t,  0.254829592f);
    p = p * t;
    // exp(-ax*ax) via native exp2: exp(x) = exp2(x * log2(e))
    const float e = __builtin_amdgcn_exp2f(-ax * ax * 1.4426950408889634f);
    const float erf_ax = __builtin_fmaf(-p, e, 1.0f);
    const float erf_s  = copysignf(erf_ax, s);
    return 0.5f * z * (1.0f + erf_s);
}

// One wave per output point (b, m). The wave processes K=32 neighbors as two
// 16-row tiles. All matmuls are done TRANSPOSED so gathered data lands in the
// B operand and constant weights in the A operand:
//   layer1: h^T  tile t = W1^T[:,4] (A, 16x4) x agg^T (B, 4x16)   -> 4 WMMA
//   layer2: kv^T chunk j = W2^T[:,4] (A, 16x4) x h^T   (B, 4x16)  -> 16 WMMA
// D layout: lane L<16 : VGPR v = kval[row=L,   chan=v]
//           lane L>=16: VGPR v = kval[row=L-16, chan=v+8]
__global__ __launch_bounds__(256)
void IntegralTransformBatch_kernel(const float* __restrict__ y,
                                   const float* __restrict__ x,
                                   const float* __restrict__ f_y,
                                   const float* __restrict__ wts,
                                   const float* __restrict__ W1,
                                   const float* __restrict__ b1,
                                   const float* __restrict__ W2,
                                   const float* __restrict__ b2,
                                   const int*   __restrict__ nbr,
                                   float*       __restrict__ out)
{
    const int tid  = threadIdx.x;
    const int wave = tid >> 5;
    const int lane = tid & 31;
    const int row  = lane & 15;        // neighbor row within tile / matrix col
    const bool hi  = lane >= 16;
    const int hoff = hi ? 8 : 0;

    const int bm = blockIdx.x * 8 + wave;
    if (bm >= B_ * M_) return;         // wave-uniform: EXEC stays all-ones
    const int b = bm / M_;
    const int m = bm % M_;

    // self point coordinates (feed high half of layer-1 B operand)
    const float xm0 = x[((long)b * M_ + m) * 2 + 0];
    const float xm1 = x[((long)b * M_ + m) * 2 + 1];

    // ---- constant operands, loaded once per wave ----
    // Layer 1 A operand: A1[t][n,k] = W1[k, 16t+n]   (W1 row-major [4][64])
    v2f w1a[4];
#pragma unroll
    for (int t = 0; t < 4; ++t) {
        const int col = 16 * t + row;
        w1a[t][0] = W1[(hi ? 2 : 0) * HID_ + col];
        w1a[t][1] = W1[(hi ? 3 : 1) * HID_ + col];
    }
    // Layer 2 A operand: A2[j][c,kk] = W2[4j+kk, c]  (W2 row-major [64][16])
    v2f w2a[16];
#pragma unroll
    for (int j = 0; j < 16; ++j) {
        w2a[j][0] = W2[(4 * j + (hi ? 2 : 0)) * DOUT_ + row];
        w2a[j][1] = W2[(4 * j + (hi ? 3 : 1)) * DOUT_ + row];
    }
    // Biases, hoisted out of the tile loop (lane-half dependent channels)
    float b1v[4][8];
#pragma unroll
    for (int t = 0; t < 4; ++t)
#pragma unroll
        for (int v = 0; v < 8; ++v)
            b1v[t][v] = b1[16 * t + v + hoff];
    float b2v[8];
#pragma unroll
    for (int v = 0; v < 8; ++v) b2v[v] = b2[v + hoff];

    float oacc[8];
#pragma unroll
    for (int v = 0; v < 8; ++v) oacc[v] = 0.0f;

    const long nb_base = ((long)b * M_ + m) * K_;
    const long tb      = (long)b * N_;       // table base for y / f_y / weights

    #pragma unroll
    for (int tt = 0; tt < 2; ++tt) {
        // ---- gather this lane's neighbor row ----
        const int  raw   = nbr[nb_base + tt * 16 + row];
        const bool valid = (raw != -1);
        const int  idx   = valid ? raw : 0;
        const float wv   = valid ? wts[tb + idx] : 0.0f;
        const float y0   = y[(tb + idx) * 2 + 0];
        const float y1   = y[(tb + idx) * 2 + 1];

        // layer-1 B operand: agg^T (4x16): rows 0,1 = gathered y (low lanes),
        // rows 2,3 = self x (high lanes)
        v2f bop;
        bop[0] = hi ? xm0 : y0;
        bop[1] = hi ? xm1 : y1;

        // ---- layer 1: h^T = W1^T x agg^T, + b1, GELU ----
        float hreg[4][8];
#pragma unroll
        for (int t = 0; t < 4; ++t) {
            v8f acc = {0.f, 0.f, 0.f, 0.f, 0.f, 0.f, 0.f, 0.f};
            acc = wmma_f32_4(w1a[t], bop, acc);
#pragma unroll
            for (int v = 0; v < 8; ++v) {
                hreg[t][v] = gelu_exact(acc[v] + b1v[t][v]);
            }
        }

        // ---- layer 2: kv^T = sum_j W2^T-chunk_j x h^T-chunk_j ----
        v8f acc2 = {0.f, 0.f, 0.f, 0.f, 0.f, 0.f, 0.f, 0.f};
#pragma unroll
        for (int t = 0; t < 4; ++t) {
            float sw[8];
#pragma unroll
            for (int v = 0; v < 8; ++v)
                sw[v] = swz<SWZ_XOR16>(hreg[t][v]);   // half-wave swap

            v2f bj;
            // chunk j = 4t+0 : K = 16t + {0,1 | 2,3}
            bj[0] = hi ? sw[2] : hreg[t][0];
            bj[1] = hi ? sw[3] : hreg[t][1];
            acc2 = wmma_f32_4(w2a[4 * t + 0], bj, acc2);
            // chunk j = 4t+1 : K = 16t + {4,5 | 6,7}
            bj[0] = hi ? sw[6] : hreg[t][4];
            bj[1] = hi ? sw[7] : hreg[t][5];
            acc2 = wmma_f32_4(w2a[4 * t + 1], bj, acc2);
            // chunk j = 4t+2 : K = 16t + {8,9 | 10,11}
            bj[0] = hi ? hreg[t][2] : sw[0];
            bj[1] = hi ? hreg[t][3] : sw[1];
            acc2 = wmma_f32_4(w2a[4 * t + 2], bj, acc2);
            // chunk j = 4t+3 : K = 16t + {12,13 | 14,15}
            bj[0] = hi ? hreg[t][6] : sw[4];
            bj[1] = hi ? hreg[t][7] : sw[5];
            acc2 = wmma_f32_4(w2a[4 * t + 3], bj, acc2);
        }

        // ---- + b2, * f_y[idx] * w * valid, accumulate over neighbors ----
        const float* fy = f_y + (tb + idx) * DOUT_;
#pragma unroll
        for (int v = 0; v < 8; ++v) {
            oacc[v] += (acc2[v] + b2v[v]) * fy[v + hoff] * wv;
        }
    }

    // ---- reduce over the 16 neighbor rows (lanes within each half-wave) ----
#pragma unroll
    for (int v = 0; v < 8; ++v) {
        float s = oacc[v];
        s += swz<SWZ_XOR1>(s);
        s += swz<SWZ_XOR2>(s);
        s += swz<SWZ_XOR4>(s);
        s += swz<SWZ_XOR8>(s);
        oacc[v] = s;
    }

    // lane 0 writes channels 0..7, lane 16 writes channels 8..15
    if (row == 0) {
        float* po = out + ((long)b * M_ + m) * DOUT_ + hoff;
#pragma unroll
        for (int v = 0; v < 8; ++v) po[v] = oacc[v];
    }
}

extern "C" void kernel_launch(void* const* d_in, const int* in_sizes, int n_in,
                              void* d_out, int out_size, void* d_ws, size_t ws_size,
                              hipStream_t stream) {
    const float* y   = (const float*)d_in[0];
    const float* x   = (const float*)d_in[1];
    const float* f_y = (const float*)d_in[2];
    const float* wts = (const float*)d_in[3];
    const float* W1  = (const float*)d_in[4];
    const float* b1  = (const float*)d_in[5];
    const float* W2  = (const float*)d_in[6];
    const float* b2  = (const float*)d_in[7];
    const int*   nbr = (const int*)d_in[8];
    float* out = (float*)d_out;

    const int total  = B_ * M_;            // one wave per (b, m)
    const int blocks = (total + 7) / 8;    // 8 waves (256 threads) per block
    IntegralTransformBatch_kernel<<<blocks, 256, 0, stream>>>(
        y, x, f_y, wts, W1, b1, W2, b2, nbr, out);
}